// ScaledPLLayer_40132174414418
// MI455X (gfx1250) — compile-verified
//
#include <hip/hip_runtime.h>
#include <hip/hip_bf16.h>

#define T_RES 100
#define K_MAX 2
#define N_PH  128
#define HW_N  4096

typedef unsigned int u32x4 __attribute__((ext_vector_type(4)));
typedef int          i32x8 __attribute__((ext_vector_type(8)));
typedef int          i32x4 __attribute__((ext_vector_type(4)));

// One workgroup per sample b. 128 threads = 4 wave32s.
// Phase 0: TDM DMA of row x[b, 0:4096] (16 KB) into LDS (single tensor_load_to_lds).
// Phase 1: gather birth/death values from LDS, compact into s_bd / s_dim.
// Phase 2: lane t evaluates 128 tent functions, keeps running top-2 per dim.
__global__ __launch_bounds__(128) void pl_landscape_kernel(
    const float* __restrict__ x,        // [B, HW]
    const int*   __restrict__ bloc,     // [B, N_PH]
    const int*   __restrict__ dloc,     // [B, N_PH]
    const int*   __restrict__ didx,     // [B, N_PH]
    float*       __restrict__ out)      // [B, 2, T_RES, K_MAX]
{
    __shared__ float  lds_x[HW_N];      // 16 KB staged row
    __shared__ float2 s_bd[N_PH];       // (birth, death) per pair
    __shared__ int    s_dim[N_PH];

    const int b   = blockIdx.x;
    const int tid = threadIdx.x;

    // ---- Phase 0: Tensor Data Mover load of x[b,:] into LDS ----------------
    // Issued by wave 0 only (TDM ignores EXEC; gate at branch granularity).
    if (tid < 32) {
        const unsigned long long ga =
            (unsigned long long)(size_t)(x + (size_t)b * HW_N);
        const unsigned lds_off = (unsigned)(size_t)lds_x;  // flat LDS addr low 32b = LDS byte offset

        // D# group 0 (ISA 8.3): count=1 | lds_addr | global_addr[56:0] | type=2
        u32x4 g0;
        g0[0] = 1u;                                   // count=1, is_restore=0, gather_mode=0
        g0[1] = lds_off;                              // lds_addr (bytes)
        g0[2] = (unsigned)(ga & 0xFFFFFFFFull);       // global_addr[31:0]
        g0[3] = (unsigned)((ga >> 32) & 0x01FFFFFFull) | 0x80000000u;  // addr[56:32], type=2

        // D# group 1 (ISA 8.4): data_size=4B, dims/tiles for a 4096x1 tile
        i32x8 g1;
        g1[0] = 0x00020000;                           // wg_mask=0, data_size=2 (4 bytes)
        g1[1] = (int)(((unsigned)HW_N & 0xFFFFu) << 16);            // tensor_dim0 lo16
        g1[2] = (int)((((unsigned)HW_N >> 16) & 0xFFFFu) | (1u << 16)); // dim0 hi16 | tensor_dim1 lo16
        g1[3] = (int)(((unsigned)HW_N & 0xFFFFu) << 16);            // tensor_dim1 hi16=0 | tile_dim0=4096
        g1[4] = 1;                                     // tile_dim1=1, tile_dim2=0
        g1[5] = HW_N;                                  // tensor_dim0_stride lo32
        g1[6] = 0;                                     // stride0 hi16 | stride1 lo16
        g1[7] = 0;                                     // stride1 hi32

        i32x4 gz4;
        gz4[0] = 0; gz4[1] = 0; gz4[2] = 0; gz4[3] = 0;  // groups 2/3 unused (<=2D tensor)
        i32x8 gz8;
        gz8[0] = 0; gz8[1] = 0; gz8[2] = 0; gz8[3] = 0;
        gz8[4] = 0; gz8[5] = 0; gz8[6] = 0; gz8[7] = 0;  // trailing group (clang-23 6-arg form)

        __builtin_amdgcn_tensor_load_to_lds(g0, g1, gz4, gz4, gz8, 0);
        __builtin_amdgcn_s_wait_tensorcnt(0);          // DMA complete for this wave
    }
    __syncthreads();                                   // publish LDS to all 4 waves

    // ---- Phase 1: gather persistence pair values from the staged row -------
    {
        const int gi = b * N_PH + tid;                 // blockDim.x == N_PH
        const int bl = bloc[gi];
        const int dl = dloc[gi];
        float2 bd;
        bd.x = lds_x[bl];                              // birth
        bd.y = lds_x[dl];                              // death
        s_bd[tid]  = bd;
        s_dim[tid] = didx[gi];
    }
    __syncthreads();

    // ---- Phase 2: tent evaluation + running top-2 per homology dim ---------
    if (tid < T_RES) {
        const float t = (float)tid * (1.0f / (float)(T_RES - 1));
        // init -1 mirrors reference's where(mask, fab, -1) before top_k
        float a0 = -1.0f, a1 = -1.0f;   // dim 0: top-1, top-2
        float c0 = -1.0f, c1 = -1.0f;   // dim 1: top-1, top-2
#pragma unroll 4
        for (int n = 0; n < N_PH; ++n) {
            const float2 bd = s_bd[n];                 // ds_load_b64 broadcast
            float fab = fminf(t - bd.x, bd.y - t);
            fab = fmaxf(fab, 0.0f);
            const bool is0 = (s_dim[n] == 0);
            const float v0 = is0 ? fab : -1.0f;
            const float v1 = is0 ? -1.0f : fab;
            // branch-free top-2 insert
            float hi = fmaxf(v0, a0);
            float lo = fminf(v0, a0);
            a0 = hi; a1 = fmaxf(a1, lo);
            hi = fmaxf(v1, c0);
            lo = fminf(v1, c0);
            c0 = hi; c1 = fmaxf(c1, lo);
        }
        // final clamp matches reference's maximum(vals, 0)
        const int base0 = ((b * 2 + 0) * T_RES + tid) * K_MAX;
        const int base1 = ((b * 2 + 1) * T_RES + tid) * K_MAX;
        out[base0 + 0] = fmaxf(a0, 0.0f);
        out[base0 + 1] = fmaxf(a1, 0.0f);
        out[base1 + 0] = fmaxf(c0, 0.0f);
        out[base1 + 1] = fmaxf(c1, 0.0f);
    }
}

extern "C" void kernel_launch(void* const* d_in, const int* in_sizes, int n_in,
                              void* d_out, int out_size, void* d_ws, size_t ws_size,
                              hipStream_t stream) {
    const float* x    = (const float*)d_in[0];
    const int*   bloc = (const int*)d_in[1];
    const int*   dloc = (const int*)d_in[2];
    const int*   didx = (const int*)d_in[3];
    float*       out  = (float*)d_out;

    const int B = in_sizes[0] / HW_N;   // 64
    pl_landscape_kernel<<<dim3(B), dim3(N_PH), 0, stream>>>(x, bloc, dloc, didx, out);
}